// RGMMNet_86423331930639
// MI455X (gfx1250) — compile-verified
//
#include <hip/hip_runtime.h>
#include <math.h>

// Problem constants (match reference).
constexpr int BB = 2;
constexpr int NN = 50000;
constexpr int FF = 8;
constexpr int EE = 800000;
constexpr int DD = 3;
constexpr int KK = 8;
constexpr int HH = 64;
constexpr int FB = FF * BB;       // 16
constexpr int NK = NN * KK;       // 400000
constexpr int CDIM = FF * KK;     // 64  (GEMM K dimension)
constexpr int MROWS = BB * NN;    // 100000
constexpr int MTILES = MROWS / 16;  // 6250
constexpr int NTILES = HH / 16;     // 4
constexpr float EPSF = 1e-14f;

typedef __attribute__((ext_vector_type(2))) float v2f;
typedef __attribute__((ext_vector_type(8))) float v8f;

// Monotone float<->uint key so atomicMax(uint) implements float max.
// key(0x0...) == 0 is below every real key, so zero-init acts as -inf.
__device__ __forceinline__ unsigned fkey(float f) {
  unsigned b = __float_as_uint(f);
  return (b & 0x80000000u) ? ~b : (b | 0x80000000u);
}
__device__ __forceinline__ float finv(unsigned k) {
  return __uint_as_float((k & 0x80000000u) ? (k & 0x7fffffffu) : ~k);
}

// ---------------- workspace init (keys, denom, agg all zero) ----------------
__global__ void k_init(unsigned* ws, int n) {
  int i = blockIdx.x * blockDim.x + threadIdx.x;
  int stride = gridDim.x * blockDim.x;
  for (; i < n; i += stride) ws[i] = 0u;
}

// ---------------- pass 1: per-edge logits -> segment max ----------------
__global__ void k_logit_max(const float* __restrict__ uval,
                            const float* __restrict__ mu,
                            const float* __restrict__ sigma,
                            const int* __restrict__ urows,
                            unsigned* __restrict__ mkey) {
  int e = blockIdx.x * blockDim.x + threadIdx.x;
  if (e >= EE) return;
  float mu_l[KK * DD], iv_l[KK * DD];
#pragma unroll
  for (int i = 0; i < KK * DD; ++i) {
    mu_l[i] = mu[i];
    float sg = sigma[i];
    iv_l[i] = 1.0f / (sg * sg + EPSF);
  }
  float uv[DD];
#pragma unroll
  for (int d = 0; d < DD; ++d) uv[d] = uval[e * DD + d];
  int row = urows[e];
#pragma unroll
  for (int k = 0; k < KK; ++k) {
    float s = 0.0f;
#pragma unroll
    for (int d = 0; d < DD; ++d) {
      float df = uv[d] - mu_l[k * DD + d];
      s += df * df * iv_l[k * DD + d];
    }
    atomicMax(&mkey[row * KK + k], fkey(-0.5f * s));
  }
}

// ---------------- pass 2: exp(logit - max) -> segment sum ----------------
__global__ void k_denom(const float* __restrict__ uval,
                        const float* __restrict__ mu,
                        const float* __restrict__ sigma,
                        const int* __restrict__ urows,
                        const unsigned* __restrict__ mkey,
                        float* __restrict__ denom) {
  int e = blockIdx.x * blockDim.x + threadIdx.x;
  if (e >= EE) return;
  float mu_l[KK * DD], iv_l[KK * DD];
#pragma unroll
  for (int i = 0; i < KK * DD; ++i) {
    mu_l[i] = mu[i];
    float sg = sigma[i];
    iv_l[i] = 1.0f / (sg * sg + EPSF);
  }
  float uv[DD];
#pragma unroll
  for (int d = 0; d < DD; ++d) uv[d] = uval[e * DD + d];
  int row = urows[e];
#pragma unroll
  for (int k = 0; k < KK; ++k) {
    float s = 0.0f;
#pragma unroll
    for (int d = 0; d < DD; ++d) {
      float df = uv[d] - mu_l[k * DD + d];
      s += df * df * iv_l[k * DD + d];
    }
    float l = -0.5f * s;
    float mval = finv(mkey[row * KK + k]);
    atomicAdd(&denom[row * KK + k], __expf(l - mval));
  }
}

// ---------------- pass 3: w * X_t[col] scattered into agg[N,K,FB] ----------
// One thread per (edge, k); k is the fast index so 8 adjacent lanes share one
// edge's X rows (two contiguous 32B runs) through L1/L2.
__global__ void k_scatter(const float* __restrict__ X,
                          const float* __restrict__ uval,
                          const float* __restrict__ mu,
                          const float* __restrict__ sigma,
                          const int* __restrict__ urows,
                          const int* __restrict__ ucols,
                          const unsigned* __restrict__ mkey,
                          const float* __restrict__ denom,
                          float* __restrict__ agg) {
  int t = blockIdx.x * blockDim.x + threadIdx.x;
  if (t >= EE * KK) return;
  int e = t >> 3;
  int k = t & (KK - 1);
  int row = urows[e];
  int col = ucols[e];
  float s = 0.0f;
#pragma unroll
  for (int d = 0; d < DD; ++d) {
    float sg = sigma[k * DD + d];
    float df = uval[e * DD + d] - mu[k * DD + d];
    s += df * df / (sg * sg + EPSF);
  }
  float l = -0.5f * s;
  float w = __expf(l - finv(mkey[row * KK + k])) / denom[row * KK + k];
  // X_t[col, f*B+b] = X[b, col, f]
  const float* x0 = X + (size_t)col * FF;                     // b = 0
  const float* x1 = X + (size_t)NN * FF + (size_t)col * FF;   // b = 1
  float* dst = agg + (size_t)row * (KK * FB) + k * FB;
#pragma unroll
  for (int f = 0; f < FF; ++f) {
    atomicAdd(dst + f * 2 + 0, w * x0[f]);
    atomicAdd(dst + f * 2 + 1, w * x1[f]);
  }
}

// ---------------- pass 4: feats[B*N, 64] @ GW[64, 64] via f32 WMMA ----------
// feats[r, c] with r = b*N + n, c = f*K + k  ==  agg[n*128 + k*16 + f*2 + b].
// One wave per 16x16 output tile; K-loop of 16 x V_WMMA_F32_16X16X4_F32.
__global__ void k_gemm(const float* __restrict__ agg,
                       const float* __restrict__ GW,
                       float* __restrict__ out) {
  int wid = blockIdx.x * (blockDim.x >> 5) + (threadIdx.x >> 5);
  if (wid >= MTILES * NTILES) return;
  int lane = threadIdx.x & 31;
  int half = lane >> 4;       // 0: K=0,1 slice; 1: K=2,3 slice (A & B frags)
  int lm = lane & 15;
  int mt = wid >> 2;
  int nt = wid & 3;
  int row0 = mt * 16;
  int col0 = nt * 16;
  int b = row0 / NN;          // N % 16 == 0, so each tile lies in one batch
  int n0 = row0 % NN;

  v8f acc = {};
#if __has_builtin(__builtin_amdgcn_wmma_f32_16x16x4_f32)
  const float* arow = agg + (size_t)(n0 + lm) * (KK * FB) + b;
#pragma unroll
  for (int kk = 0; kk < CDIM / 4; ++kk) {
    int c0 = kk * 4 + half * 2;
    int c1 = c0 + 1;
    v2f a, bf;
    // A fragment: lanes 0-15 -> K=c0,c0+1 of row (n0+lm); lanes 16-31 -> K+2
    a.x = arow[(c0 & 7) * FB + ((c0 >> 3) << 1)];
    a.y = arow[(c1 & 7) * FB + ((c1 >> 3) << 1)];
    // B fragment mirrors A: lane holds column (col0+lm), rows c0,c0+1
    bf.x = GW[c0 * HH + col0 + lm];
    bf.y = GW[c1 * HH + col0 + lm];
    acc = __builtin_amdgcn_wmma_f32_16x16x4_f32(
        /*neg_a=*/false, a, /*neg_b=*/false, bf,
        /*c_mod=*/(short)0, acc, /*reuse_a=*/false, /*reuse_b=*/false);
  }
  // C/D layout: VGPR j holds M=j (lanes 0-15) and M=j+8 (lanes 16-31), N=lm.
#pragma unroll
  for (int j = 0; j < 8; ++j) {
    int r = row0 + half * 8 + j;
    out[(size_t)r * HH + col0 + lm] = acc[j];
  }
#else
  // Scalar fallback (keeps this file compiling if the builtin is absent).
#pragma unroll
  for (int j = 0; j < 8; ++j) {
    int r = row0 + half * 8 + j;
    int n = r % NN;
    int bb = r / NN;
    float s = 0.0f;
    for (int c = 0; c < CDIM; ++c) {
      float fv = agg[(size_t)n * (KK * FB) + (c & 7) * FB + ((c >> 3) << 1) + bb];
      s += fv * GW[c * HH + col0 + lm];
    }
    out[(size_t)r * HH + col0 + lm] = s;
  }
#endif
}

extern "C" void kernel_launch(void* const* d_in, const int* in_sizes, int n_in,
                              void* d_out, int out_size, void* d_ws, size_t ws_size,
                              hipStream_t stream) {
  const float* X     = (const float*)d_in[0];  // [B, N, F]
  const float* uval  = (const float*)d_in[1];  // [E, D]
  const float* mu    = (const float*)d_in[2];  // [K, D]
  const float* sigma = (const float*)d_in[3];  // [K, D]
  const float* GW    = (const float*)d_in[4];  // [F*K, H]
  const int* urows   = (const int*)d_in[5];    // [E]
  const int* ucols   = (const int*)d_in[6];    // [E]
  float* out = (float*)d_out;                  // [B, N, H]

  // Workspace layout (words): [ mkey: NK u32 | denom: NK f32 | agg: NK*FB f32 ]
  unsigned* mkey = (unsigned*)d_ws;
  float* denom   = (float*)d_ws + NK;
  float* agg     = (float*)d_ws + 2 * NK;
  const int ws_words = 2 * NK + NK * FB;  // 7.2M words = 28.8 MB

  k_init<<<2048, 256, 0, stream>>>((unsigned*)d_ws, ws_words);

  k_logit_max<<<(EE + 255) / 256, 256, 0, stream>>>(uval, mu, sigma, urows, mkey);

  k_denom<<<(EE + 255) / 256, 256, 0, stream>>>(uval, mu, sigma, urows, mkey, denom);

  k_scatter<<<(EE * KK + 255) / 256, 256, 0, stream>>>(
      X, uval, mu, sigma, urows, ucols, mkey, denom, agg);

  int waves = MTILES * NTILES;                 // 25000 waves, 8 waves/block
  int blocks = (waves + 7) / 8;
  k_gemm<<<blocks, 256, 0, stream>>>(agg, GW, out);
}